// GRAPH_VAE_39247411151511
// MI455X (gfx1250) — compile-verified
//
#include <hip/hip_runtime.h>

typedef __attribute__((ext_vector_type(2))) float v2f;
typedef __attribute__((ext_vector_type(8))) float v8f;

static constexpr int kNodes  = 100000;
static constexpr int kEdges  = 3200000;
static constexpr int kGraphs = 64;
static constexpr int kLatent = 64;
static constexpr int kFcOut  = 160000;

// ----------------------------- degree / norm --------------------------------
__global__ void deg_init_k(float* __restrict__ deg) {
  int i = blockIdx.x * blockDim.x + threadIdx.x;
  if (i < kNodes) deg[i] = 1.0f;  // self-loop
}
__global__ void deg_edge_k(const int* __restrict__ dst, float* __restrict__ deg) {
  int e = blockIdx.x * blockDim.x + threadIdx.x;
  if (e < kEdges) atomicAdd(&deg[dst[e]], 1.0f);
}
__global__ void deg_rsqrt_k(float* __restrict__ deg) {
  int i = blockIdx.x * blockDim.x + threadIdx.x;
  if (i < kNodes) deg[i] = rsqrtf(deg[i]);
}

// ------------------- dense transform h = in @ W (WMMA f32) ------------------
// A 16x4 layout: lane<16 holds row M=lane, VGPR0=K+0, VGPR1=K+1; lane>=16 holds
// the same row with K+2/K+3.  B 4x16: VGPR0 = rows K+0 (lanes 0-15) / K+2
// (lanes 16-31); VGPR1 = K+1 / K+3.  C/D: VGPR v -> row v (lanes 0-15) or v+8.
template<int DIN, int DOUT>
__global__ void gemm_wmma_k(const float* __restrict__ in, const float* __restrict__ W,
                            float* __restrict__ h) {
  constexpr int NT = DOUT / 16;
  const int lane = threadIdx.x & 31;
  const int wave = threadIdx.x >> 5;
  const int tile = blockIdx.x * (blockDim.x >> 5) + wave;
  const int base = tile * 16;
  if (base >= kNodes) return;                 // uniform per-wave exit
  const int half = lane >> 4;
  const int l15  = lane & 15;
  int arow = base + l15;
  if (arow >= kNodes) arow = kNodes - 1;      // clamp keeps EXEC all-ones
  const float* arp = in + (size_t)arow * DIN + half * 2;

  v8f acc[NT] = {};
#pragma unroll
  for (int k = 0; k < DIN; k += 4) {
    const float2 av = *(const float2*)(arp + k);
    v2f a; a.x = av.x; a.y = av.y;
#pragma unroll
    for (int t = 0; t < NT; ++t) {
      v2f b;
      b.x = W[(k + half * 2    ) * DOUT + t * 16 + l15];
      b.y = W[(k + half * 2 + 1) * DOUT + t * 16 + l15];
      acc[t] = __builtin_amdgcn_wmma_f32_16x16x4_f32(false, a, false, b,
                                                     (short)0, acc[t], false, false);
    }
  }
#pragma unroll
  for (int t = 0; t < NT; ++t) {
#pragma unroll
    for (int v = 0; v < 8; ++v) {
      int row = base + v + half * 8;
      if (row < kNodes) h[(size_t)row * DOUT + t * 16 + l15] = acc[t][v];
    }
  }
}

// -------------------- agg = h*dinv^2 + bias (self loop) ---------------------
template<int DIM>
__global__ void init_agg_k(const float* __restrict__ h, const float* __restrict__ dinv,
                           const float* __restrict__ bias, float* __restrict__ agg) {
  int i = blockIdx.x * blockDim.x + threadIdx.x;
  if (i >= kNodes * DIM) return;
  int node = i / DIM;
  int c    = i - node * DIM;
  float di = dinv[node];
  agg[i] = h[i] * di * di + bias[c];
}

// -------------------- edge scatter: agg[dst] += coef*h[src] -----------------
template<int DIM>
__global__ void edge_scatter_k(const int* __restrict__ src, const int* __restrict__ dst,
                               const float* __restrict__ h, const float* __restrict__ dinv,
                               float* __restrict__ agg) {
  int e = blockIdx.x * blockDim.x + threadIdx.x;
  if (e >= kEdges) return;
  int s = src[e], d = dst[e];
  float coef = dinv[s] * dinv[d];
  const float4* hp = (const float4*)(h + (size_t)s * DIM);
  float* ap = agg + (size_t)d * DIM;
#pragma unroll
  for (int c = 0; c < DIM / 4; ++c) {
    float4 hv = hp[c];
    atomicAdd(ap + 4 * c + 0, hv.x * coef);
    atomicAdd(ap + 4 * c + 1, hv.y * coef);
    atomicAdd(ap + 4 * c + 2, hv.z * coef);
    atomicAdd(ap + 4 * c + 3, hv.w * coef);
  }
}

__global__ void relu_k(float* __restrict__ a, int n) {
  int i = blockIdx.x * blockDim.x + threadIdx.x;
  if (i < n) a[i] = fmaxf(a[i], 0.0f);
}

// ------------------------------- mean pool ----------------------------------
__global__ void pool_zero_k(float* __restrict__ pool, float* __restrict__ cnt) {
  int i = blockIdx.x * blockDim.x + threadIdx.x;
  if (i < kGraphs * kLatent) pool[i] = 0.0f;
  if (i < kGraphs) cnt[i] = 0.0f;
}
__global__ void pool_acc_k(const float* __restrict__ feat, const int* __restrict__ batch,
                           float* __restrict__ pool, float* __restrict__ cnt) {
  int i = blockIdx.x * blockDim.x + threadIdx.x;
  if (i >= kNodes) return;
  int g = batch[i];
  atomicAdd(&cnt[g], 1.0f);
  const float* fp = feat + (size_t)i * kLatent;
  float* pp = pool + (size_t)g * kLatent;
#pragma unroll 8
  for (int c = 0; c < kLatent; ++c) atomicAdd(pp + c, fp[c]);
}
__global__ void pool_final_k(const float* __restrict__ pool, const float* __restrict__ cnt,
                             float* __restrict__ out_global, float* __restrict__ out_ws) {
  int i = blockIdx.x * blockDim.x + threadIdx.x;
  if (i >= kGraphs * kLatent) return;
  float m = pool[i] / fmaxf(cnt[i / kLatent], 1.0f);
  out_global[i] = m;
  out_ws[i] = m;
}

// --------------------------- reparameterization -----------------------------
__global__ void z_k(const float* __restrict__ mu, const float* __restrict__ logv,
                    const float* __restrict__ eps, float* __restrict__ z) {
  int i = blockIdx.x * blockDim.x + threadIdx.x;
  if (i < kGraphs * kLatent) z[i] = mu[i] + eps[i] * expf(0.5f * logv[i]);
}

// --------------- decoder: sigmoid(z @ Wfc + bfc) via WMMA f32 ---------------
__global__ void decoder_wmma_k(const float* __restrict__ z, const float* __restrict__ Wfc,
                               const float* __restrict__ bfc, float* __restrict__ out) {
  __shared__ float zs[kGraphs * kLatent];  // 16 KB, stays hot for all 8 waves
  for (int i = threadIdx.x; i < kGraphs * kLatent; i += blockDim.x) zs[i] = z[i];
  __syncthreads();

  const int lane = threadIdx.x & 31;
  const int wave = threadIdx.x >> 5;
  const int tile = blockIdx.x * (blockDim.x >> 5) + wave;  // 16-col tile of Wfc
  if (tile * 16 >= kFcOut) return;
  const int half = lane >> 4;
  const int l15  = lane & 15;
  const int col  = tile * 16 + l15;

  v8f acc[4] = {};
  for (int k = 0; k < kLatent; k += 4) {
    v2f b;
    b.x = Wfc[(size_t)(k + half * 2    ) * kFcOut + col];
    b.y = Wfc[(size_t)(k + half * 2 + 1) * kFcOut + col];
#pragma unroll
    for (int m = 0; m < 4; ++m) {
      const float2 av = *(const float2*)(&zs[(m * 16 + l15) * kLatent + k + half * 2]);
      v2f a; a.x = av.x; a.y = av.y;
      acc[m] = __builtin_amdgcn_wmma_f32_16x16x4_f32(false, a, false, b,
                                                     (short)0, acc[m], false, false);
    }
  }
  const float bias = bfc[col];
#pragma unroll
  for (int m = 0; m < 4; ++m) {
#pragma unroll
    for (int v = 0; v < 8; ++v) {
      int row = m * 16 + v + half * 8;
      float s = acc[m][v] + bias;
      out[(size_t)row * kFcOut + col] = 1.0f / (1.0f + __expf(-s));
    }
  }
}

// ------------------------------- launcher -----------------------------------
extern "C" void kernel_launch(void* const* d_in, const int* in_sizes, int n_in,
                              void* d_out, int out_size, void* d_ws, size_t ws_size,
                              hipStream_t stream) {
  (void)in_sizes; (void)n_in; (void)out_size; (void)ws_size;
  const float* x    = (const float*)d_in[0];
  const int*  eidx  = (const int*)d_in[1];
  const int*  batch = (const int*)d_in[2];
  const float* W1m = (const float*)d_in[3];  const float* b1m = (const float*)d_in[4];
  const float* W2m = (const float*)d_in[5];  const float* b2m = (const float*)d_in[6];
  const float* W3m = (const float*)d_in[7];  const float* b3m = (const float*)d_in[8];
  const float* W1l = (const float*)d_in[9];  const float* b1l = (const float*)d_in[10];
  const float* W2l = (const float*)d_in[11]; const float* b2l = (const float*)d_in[12];
  const float* W3l = (const float*)d_in[13]; const float* b3l = (const float*)d_in[14];
  const float* Wfc = (const float*)d_in[15]; const float* bfc = (const float*)d_in[16];
  const float* eps = (const float*)d_in[17];

  const int* src = eidx;
  const int* dst = eidx + kEdges;

  float* ws   = (float*)d_ws;
  float* dinv = ws;                          // 100000
  float* A    = ws + kNodes;                 // h buffer   (100000*64)
  float* B    = A + (size_t)kNodes * 64;     // agg/input  (100000*64)
  float* pool = B + (size_t)kNodes * 64;     // 64*64
  float* cnt  = pool + kGraphs * kLatent;    // 64
  float* muW  = cnt + kGraphs;               // 64*64
  float* lvW  = muW + kGraphs * kLatent;     // 64*64
  float* zb   = lvW + kGraphs * kLatent;     // 64*64

  float* out    = (float*)d_out;
  float* mu_out = out + (size_t)kGraphs * kFcOut;
  float* lv_out = mu_out + kGraphs * kLatent;

  const int T = 256;
  auto cdiv = [](long a, long b) { return (int)((a + b - 1) / b); };

  deg_init_k<<<cdiv(kNodes, T), T, 0, stream>>>(dinv);
  deg_edge_k<<<cdiv(kEdges, T), T, 0, stream>>>(dst, dinv);
  deg_rsqrt_k<<<cdiv(kNodes, T), T, 0, stream>>>(dinv);

  const int gemm_blocks = cdiv(cdiv(kNodes, 16), T / 32);

  auto run_branch = [&](const float* Wa, const float* ba, const float* Wb, const float* bb,
                        const float* Wc, const float* bc, float* out_g, float* out_w) {
    gemm_wmma_k<64, 16><<<gemm_blocks, T, 0, stream>>>(x, Wa, A);
    init_agg_k<16><<<cdiv((long)kNodes * 16, T), T, 0, stream>>>(A, dinv, ba, B);
    edge_scatter_k<16><<<cdiv(kEdges, T), T, 0, stream>>>(src, dst, A, dinv, B);
    relu_k<<<cdiv((long)kNodes * 16, T), T, 0, stream>>>(B, kNodes * 16);

    gemm_wmma_k<16, 32><<<gemm_blocks, T, 0, stream>>>(B, Wb, A);
    init_agg_k<32><<<cdiv((long)kNodes * 32, T), T, 0, stream>>>(A, dinv, bb, B);
    edge_scatter_k<32><<<cdiv(kEdges, T), T, 0, stream>>>(src, dst, A, dinv, B);
    relu_k<<<cdiv((long)kNodes * 32, T), T, 0, stream>>>(B, kNodes * 32);

    gemm_wmma_k<32, 64><<<gemm_blocks, T, 0, stream>>>(B, Wc, A);
    init_agg_k<64><<<cdiv((long)kNodes * 64, T), T, 0, stream>>>(A, dinv, bc, B);
    edge_scatter_k<64><<<cdiv(kEdges, T), T, 0, stream>>>(src, dst, A, dinv, B);
    relu_k<<<cdiv((long)kNodes * 64, T), T, 0, stream>>>(B, kNodes * 64);

    pool_zero_k<<<cdiv(kGraphs * kLatent, T), T, 0, stream>>>(pool, cnt);
    pool_acc_k<<<cdiv(kNodes, T), T, 0, stream>>>(B, batch, pool, cnt);
    pool_final_k<<<cdiv(kGraphs * kLatent, T), T, 0, stream>>>(pool, cnt, out_g, out_w);
  };

  run_branch(W1m, b1m, W2m, b2m, W3m, b3m, mu_out, muW);
  run_branch(W1l, b1l, W2l, b2l, W3l, b3l, lv_out, lvW);

  z_k<<<cdiv(kGraphs * kLatent, T), T, 0, stream>>>(muW, lvW, eps, zb);

  decoder_wmma_k<<<cdiv(kFcOut / 16, T / 32), T, 0, stream>>>(zb, Wfc, bfc, out);
}